// SpeedPerturb_53240414601546
// MI455X (gfx1250) — compile-verified
//
#include <hip/hip_runtime.h>
#include <math.h>

// ---------------------------------------------------------------------------
// Kaldi LinearResample 16000 -> 14400 as a polyphase GEMM on CDNA5 WMMA.
//   P = 9 phases, conv stride = 10, taps widened to a shared 23-sample
//   window [-6, +16] so that  D(16x16) = W'(16x24) x X(24x16)  via six
//   V_WMMA_F32_16X16X4_F32 accumulations. Memory-bound: ~58 MB @ 23.3 TB/s.
// ---------------------------------------------------------------------------

typedef __attribute__((ext_vector_type(2))) float v2f;
typedef __attribute__((ext_vector_type(8))) float v8f;

#define R_ORIG   16000
#define R_NEW    14400
#define T_IN     480000
#define T_OUT    432000
#define PPH      9          // phases (P)
#define CSTRIDE  10         // input stride per output period
#define EXTW     24         // extended tap window 23, padded to 24 (K dim)
#define GMIN     (-6)       // global minimum tap offset (= first_idx[0])
#define KTILE    16         // k-periods per wave tile (WMMA N)
#define WAVES    8          // waves per block
#define KBLOCK   (KTILE * WAVES)            // 128 k-periods per block
#define KTOTAL   (T_OUT / PPH)              // 48000 k-periods per batch row
#define NBLK     (KTOTAL / KBLOCK)          // 375 blocks along k
#define NSTAGE   (CSTRIDE * (KBLOCK - 1) + EXTW)  // 1294 staged floats

// ---------------------------------------------------------------------------
// Setup: build the extended 16x24 weight matrix W' once (replicates
// torchaudio/Kaldi _indices_and_weights in f64). Rows 9..15 and col 23 = 0.
// Entries outside a phase's true [min_idx, max_idx] window fall out of the
// |dt| < ww condition and evaluate to 0, so the widened matrix is exact.
// ---------------------------------------------------------------------------
__global__ void resample_build_weights(float* __restrict__ wout) {
    int i = threadIdx.x;            // 0..383  (16 * 24)
    if (i >= 16 * EXTW) return;
    int p = i / EXTW;
    int m = i % EXTW;
    double val = 0.0;
    if (p < PPH && m < EXTW - 1) {
        const double PI     = 3.14159265358979323846;
        const double cutoff = 0.99 * 0.5 * (double)R_NEW;   // 7128.0
        const double ww     = 6.0 / (2.0 * cutoff);         // lowpass window
        double out_t = (double)p / (double)R_NEW;
        double dt    = (double)(m + GMIN) / (double)R_ORIG - out_t;
        if (fabs(dt) < ww) {
            double w = 0.5 * (1.0 + cos(2.0 * PI * cutoff / 6.0 * dt));
            if (dt != 0.0) w *= sin(2.0 * PI * cutoff * dt) / (PI * dt);
            else           w *= 2.0 * cutoff;
            val = w / (double)R_ORIG;
        }
    }
    wout[i] = (float)val;
}

// ---------------------------------------------------------------------------
// Main kernel: one block = 8 waves = 128 k-periods = 1152 outputs for one
// batch row. Input window staged coalesced into LDS (zero-filled OOB handles
// the reference's validity mask, which is phase-independent). Each wave runs
// six f32 WMMAs with the weight operand held in registers.
// ---------------------------------------------------------------------------
__global__ void __launch_bounds__(256) resample_wmma_kernel(
    const float* __restrict__ x,      // (B, T_IN)
    const float* __restrict__ wmat,   // (16, 24) extended weights
    float* __restrict__ out)          // (B, T_OUT)
{
    __shared__ __align__(16) float xs[NSTAGE + 2];

    const int tid = threadIdx.x;
    const int b   = blockIdx.y;
    const int k0  = blockIdx.x * KBLOCK;

    // ---- stage input window [10*k0 - 6, 10*k0 - 6 + NSTAGE) into LDS ----
    const float* xrow    = x + (long)b * T_IN;
    const int    base_in = k0 * CSTRIDE + GMIN;
    for (int i = tid; i < NSTAGE; i += 256) {
        int g = base_in + i;
        float v = 0.0f;
        if (g >= 0 && g < T_IN) v = xrow[g];
        xs[i] = v;
    }

    const int lane = tid & 31;
    const int wv   = tid >> 5;
    const int half = lane >> 4;      // lane half selects K sub-pair / M rows
    const int sub  = lane & 15;      // A: M row, B/D: N column
    const int kh   = half * 2;       // K offset inside each K=4 chunk

    // ---- A operand (weights) resident in 12 VGPRs for the whole kernel ----
    // ISA 32-bit A 16x4 layout: lanes 0-15 hold K={0,1}, lanes 16-31 K={2,3}.
    v2f a[6];
#pragma unroll
    for (int c = 0; c < 6; ++c)
        a[c] = *(const v2f*)(wmat + sub * EXTW + 4 * c + kh);

    __syncthreads();

    // ---- six K=4 WMMA accumulations: D += W'[:,4c:4c+4] * X[4c:4c+4,:] ----
    // B[m][col] = xs[10*col + m]; per-lane pair (kh, kh+1) is contiguous and
    // 8B aligned -> float2 LDS loads, stride-10 pattern is bank-conflict-free.
    const int xoff = wv * (KTILE * CSTRIDE) + sub * CSTRIDE + kh;
    v8f acc = {};
#pragma unroll
    for (int c = 0; c < 6; ++c) {
        v2f bv = *(const v2f*)(xs + xoff + 4 * c);
        acc = __builtin_amdgcn_wmma_f32_16x16x4_f32(
            /*neg_a=*/false, a[c], /*neg_b=*/false, bv,
            /*c_mod=*/(short)0, acc, /*reuse_a=*/false, /*reuse_b=*/false);
    }

    // ---- scatter D: lanes 0-15 hold phases 0-7, lanes 16-31 hold phase 8 --
    const int kcol = k0 + wv * KTILE + sub;
    float* orow = out + (long)b * T_OUT + kcol * PPH;
    if (half == 0) {
#pragma unroll
        for (int r = 0; r < 8; ++r) orow[r] = acc[r];
    } else {
        orow[8] = acc[0];
    }
}

// ---------------------------------------------------------------------------
extern "C" void kernel_launch(void* const* d_in, const int* in_sizes, int n_in,
                              void* d_out, int out_size, void* d_ws, size_t ws_size,
                              hipStream_t stream) {
    const float* x   = (const float*)d_in[0];
    float*       out = (float*)d_out;
    float*       wmat = (float*)d_ws;          // 384 floats of scratch

    int B = in_sizes[0] / T_IN;                // 16

    hipLaunchKernelGGL(resample_build_weights, dim3(1), dim3(384), 0, stream,
                       wmat);

    dim3 grid(NBLK, B);
    hipLaunchKernelGGL(resample_wmma_kernel, grid, dim3(256), 0, stream,
                       x, wmat, out);
}